// ProbsNet_50139448214007
// MI455X (gfx1250) — compile-verified
//
#include <hip/hip_runtime.h>
#include <hip/hip_bf16.h>

typedef __attribute__((ext_vector_type(2))) float v2f;
typedef __attribute__((ext_vector_type(4))) float v4f;
typedef __attribute__((ext_vector_type(8))) float v8f;

#define FDIM 131072
#define NROW 84
#define CH   16          // chunks per row
#define TPB  256

// ---------------------------------------------------------------------------
// Kernel A: streaming sigmoid-dot partial reduction. Memory bound:
// 176 MB read-once => ~7.6 us floor at 23.3 TB/s. NT b128 loads, 8 loads
// in flight per wave per group. Deterministic fixed-order reduction.
// One block = one (row, chunk); partials layout [row][CH] per tensor.
// ---------------------------------------------------------------------------
__global__ __launch_bounds__(TPB) void rowdot_kernel(
    const float* __restrict__ ST, const float* __restrict__ W,
    const float* __restrict__ BEV, const float* __restrict__ pBEV,
    const float* __restrict__ pB, float* __restrict__ partials)
{
    const int chunk = blockIdx.x;
    const int row   = blockIdx.y;
    const float bev = pBEV[0] * BEV[0];
    const float pb  = pB[0];

    const int elemsPerChunk = FDIM / CH;         // 8192 floats
    // 2048 b128 vectors per chunk, 256 threads -> 8 vec-pairs per thread.

    const size_t base = (size_t)row * FDIM + (size_t)chunk * elemsPerChunk;
    const v4f* __restrict__ st4 = (const v4f*)(ST + base);
    const v4f* __restrict__ w4  = (const v4f*)(W  + base);

    float acc = 0.f;
    #pragma unroll
    for (int g = 0; g < 2; ++g) {
        v4f s[4], w[4];
        #pragma unroll
        for (int u = 0; u < 4; ++u) {            // 8 NT b128 loads in flight
            const int v = (g * 4 + u) * TPB + (int)threadIdx.x;
            s[u] = __builtin_nontemporal_load(&st4[v]);
            w[u] = __builtin_nontemporal_load(&w4[v]);
        }
        #pragma unroll
        for (int u = 0; u < 4; ++u) {
            acc += w[u].x / (1.f + __expf(-pb * (bev + s[u].x)));
            acc += w[u].y / (1.f + __expf(-pb * (bev + s[u].y)));
            acc += w[u].z / (1.f + __expf(-pb * (bev + s[u].z)));
            acc += w[u].w / (1.f + __expf(-pb * (bev + s[u].w)));
        }
    }

    // wave32 tree reduce (fixed order)
    #pragma unroll
    for (int off = 16; off > 0; off >>= 1)
        acc += __shfl_down(acc, off, 32);

    __shared__ float red[TPB / 32];
    const int wid = threadIdx.x >> 5, lid = threadIdx.x & 31;
    if (lid == 0) red[wid] = acc;
    __syncthreads();
    if (threadIdx.x == 0) {
        float s = 0.f;
        #pragma unroll
        for (int i = 0; i < TPB / 32; ++i) s += red[i];
        partials[row * CH + chunk] = s;
    }
}

// ---------------------------------------------------------------------------
// calc_probs entry j of the 84-vector derived from 4 softmax probs p[0..3]:
// layout per i (21 entries): [p_i, {p_i*p_j, p_i*p_j*p_k (k=0..3)} for j=0..3]
// ---------------------------------------------------------------------------
__device__ __forceinline__ float pentry(const float* p, int j) {
    const int i   = j / 21;
    int       rem = j % 21;
    const float pi = p[i];
    if (rem == 0) return pi;
    rem -= 1;
    const int jj = rem / 5;
    const int rr = rem % 5;
    const float pij = pi * p[jj];
    if (rr == 0) return pij;
    return pij * p[rr - 1];
}

// ---------------------------------------------------------------------------
// Kernel B: one wave32. Both WMMA operands are staged zero-padded to the
// full 16-row shape in LDS, so the matrix loop is branch-free:
//   2 x ds_load_b64 + v_wmma_f32_16x16x4_f32 per K-step (24 steps).
//   A16 rows 0-4 = p0..p4 (calc_probs), rows 5-15 & cols 84-95 = 0.
//   T16 row 0 = tmp0, row 1 = tmp1, rows 2-15 = 0.
//   D[0,0] = p0.tmp0 ; D[q,1] = pq.tmp1 (q=1..4)
//   out = (D[0,0] + D[1,1] + D[2,1] + D[3,1] + D[4,1]) / 5
// ---------------------------------------------------------------------------
__global__ __launch_bounds__(32) void combine_kernel(
    const float* __restrict__ partials,   // [2][NROW][CH]
    const float* __restrict__ q0, const float* __restrict__ q1,
    const float* __restrict__ q2, const float* __restrict__ q3,
    const float* __restrict__ q4, float* __restrict__ out)
{
    const int lane = threadIdx.x;

    __shared__ float A16[16][96];   // padded A operand
    __shared__ float T16[16][96];   // padded B operand
    __shared__ float P4[5][4];

    // Zero-pad + fill tmp rows (deterministic sum of 16 chunk partials).
    for (int j = lane; j < 96; j += 32) {
        #pragma unroll
        for (int r = 0; r < 16; ++r) A16[r][j] = 0.f;
        #pragma unroll
        for (int r = 2; r < 16; ++r) T16[r][j] = 0.f;
        #pragma unroll
        for (int t = 0; t < 2; ++t) {
            float s = 0.f;
            if (j < NROW) {
                const float* pp = partials + ((size_t)t * NROW + j) * CH;
                #pragma unroll
                for (int c = 0; c < CH; ++c) s += pp[c];
            }
            T16[t][j] = s;
        }
    }

    // 5 softmaxes (4 logits each).
    if (lane < 5) {
        const float* const qs[5] = {q0, q1, q2, q3, q4};
        const float* l = qs[lane];
        float m = fmaxf(fmaxf(l[0], l[1]), fmaxf(l[2], l[3]));
        float e0 = __expf(l[0] - m), e1 = __expf(l[1] - m);
        float e2 = __expf(l[2] - m), e3 = __expf(l[3] - m);
        float inv = 1.f / (e0 + e1 + e2 + e3);
        P4[lane][0] = e0 * inv; P4[lane][1] = e1 * inv;
        P4[lane][2] = e2 * inv; P4[lane][3] = e3 * inv;
    }
    __syncthreads();

    // Fill A rows 0..4; all calc_probs branching happens here, outside the
    // WMMA loop.
    for (int j = lane; j < NROW; j += 32) {
        #pragma unroll
        for (int q = 0; q < 5; ++q)
            A16[q][j] = pentry(P4[q], j);
    }
    __syncthreads();

    // ISA layouts (16x16x4 f32), identical row index for A and B operands:
    //  A: M = lane&15; a.x -> K = 4c + (lane>=16 ? 2 : 0); a.y -> K+1
    //  B: N = lane&15; b.x -> same K row;                  b.y -> K+1
    //  D: c[r] lane n = D[r, n] (lanes 0-15)
    const int r16  = lane & 15;
    const int koff = (lane >= 16) ? 2 : 0;

    v8f acc = {};
    for (int cblk = 0; cblk < 24; ++cblk) {
        const int k0 = cblk * 4 + koff;           // even -> 8B aligned
        v2f a = *(const v2f*)&A16[r16][k0];       // ds_load_b64
        v2f b = *(const v2f*)&T16[r16][k0];       // ds_load_b64
        acc = __builtin_amdgcn_wmma_f32_16x16x4_f32(
            false, a, false, b, (short)0, acc, false, false);
    }

    float r0 = __shfl(acc[0], 0, 32);   // D[0,0] = p0 . tmp0
    float r1 = __shfl(acc[1], 1, 32);   // D[1,1] = p1 . tmp1
    float r2 = __shfl(acc[2], 1, 32);
    float r3 = __shfl(acc[3], 1, 32);
    float r4 = __shfl(acc[4], 1, 32);
    if (lane == 0) out[0] = (r0 + r1 + r2 + r3 + r4) * 0.2f;
}

// ---------------------------------------------------------------------------
// Inputs (setup_inputs order):
//  0:BEV 1:ST0 2:Weight0 3:ST1 4:Weight1 5:Problem 6..10:probs0..4 11:pBEV 12:pB
// ---------------------------------------------------------------------------
extern "C" void kernel_launch(void* const* d_in, const int* in_sizes, int n_in,
                              void* d_out, int out_size, void* d_ws, size_t ws_size,
                              hipStream_t stream) {
    const float* BEV  = (const float*)d_in[0];
    const float* ST0  = (const float*)d_in[1];
    const float* W0   = (const float*)d_in[2];
    const float* ST1  = (const float*)d_in[3];
    const float* W1   = (const float*)d_in[4];
    const float* q0   = (const float*)d_in[6];
    const float* q1   = (const float*)d_in[7];
    const float* q2   = (const float*)d_in[8];
    const float* q3   = (const float*)d_in[9];
    const float* q4   = (const float*)d_in[10];
    const float* pBEV = (const float*)d_in[11];
    const float* pB   = (const float*)d_in[12];
    float* out = (float*)d_out;

    float* partials = (float*)d_ws;            // [2][NROW][CH] = 10.5 KB

    dim3 grid(CH, NROW, 1);
    rowdot_kernel<<<grid, TPB, 0, stream>>>(ST0, W0, BEV, pBEV, pB, partials);
    rowdot_kernel<<<grid, TPB, 0, stream>>>(ST1, W1, BEV, pBEV, pB,
                                            partials + (size_t)NROW * CH);
    combine_kernel<<<1, 32, 0, stream>>>(partials, q0, q1, q2, q3, q4, out);
}